// Projection_40621800685676
// MI455X (gfx1250) — compile-verified
//
#include <hip/hip_runtime.h>
#include <math.h>

// Siddon exact forward projector for MI455X (gfx1250, wave32).
// One thread per LOR; incremental 3D-DDA over a 128^3 grid.
// - LOR block staged into LDS via GLOBAL_LOAD_ASYNC_TO_LDS_B128 (ASYNCcnt)
// - global_prefetch warms the first gather
// - image (8MB) lives in the 192MB L2 after warm-up -> gathers are L2 hits
// - inner loop is branchless (clamped gather + masked weight) so the only
//   per-iteration control flow is the exit test.

#define THREADS 256
#define GRID_N  128

__device__ __forceinline__ void async_lds_load_b128(unsigned lds_byte_off,
                                                    const void* gptr) {
    unsigned long long ga = (unsigned long long)gptr;
    // VDST = per-lane LDS byte offset, VADDR = 64-bit global address (GV mode)
    asm volatile("global_load_async_to_lds_b128 %0, %1, off"
                 :: "v"(lds_byte_off), "v"(ga)
                 : "memory");
}

__device__ __forceinline__ void wait_asynccnt0() {
    asm volatile("s_wait_asynccnt 0x0" ::: "memory");
}

__global__ __launch_bounds__(THREADS)
void siddon_fp_kernel(const float* __restrict__ image,
                      const float* __restrict__ lors,
                      float* __restrict__ out, int n)
{
    __shared__ float s_lor[THREADS * 6];

    const int tid = threadIdx.x;
    const long long base_ray = (long long)blockIdx.x * THREADS;
    const long long base_dw  = base_ray * 6;

    // ---------- coalesced async stage of this block's LOR data ----------
    // 256 rays * 6 floats = 6144 B = 384 x 16B chunks, fully coalesced.
    const int CHUNKS = (THREADS * 6) / 4;                 // 384
    long long avail_dw = (long long)n * 6 - base_dw;      // >= 6 always
    int avail = (int)(avail_dw >> 2);                     // whole 16B chunks
    if (avail > CHUNKS) avail = CHUNKS;
    if (avail < 1) avail = 1;

    // low 32 bits of a flat shared pointer == workgroup LDS byte offset
    unsigned lds_base = (unsigned)(unsigned long long)(const void*)&s_lor[0];
    const float* gsrc = lors + base_dw;

    for (int c = tid; c < CHUNKS; c += THREADS) {
        int cc = (c < avail) ? c : (avail - 1);           // clamp tail (no OOB)
        async_lds_load_b128(lds_base + (unsigned)c * 16u,
                            gsrc + (size_t)cc * 4);
    }
    wait_asynccnt0();
    __syncthreads();

    const long long ray = base_ray + tid;
    if (ray >= n) return;

    // ---------- load my LOR from LDS ----------
    const float* L = &s_lor[tid * 6];
    const float p0x = L[0], p0y = L[1], p0z = L[2];
    const float dx = L[3] - p0x, dy = L[4] - p0y, dz = L[5] - p0z;

    const float EPS = 1e-9f;
    const float ORG = -1.0f;          // ORIGIN (all axes)
    const float VOX = 0.015625f;      // 2/128
    const float INV = 64.0f;          // 1/VOX

    const float dsx = (fabsf(dx) < EPS) ? EPS : dx;
    const float dsy = (fabsf(dy) < EPS) ? EPS : dy;
    const float dsz = (fabsf(dz) < EPS) ? EPS : dz;

    // slab clip: alpha range inside [-1,1]^3, clamped to [0,1]
    const float ax0 = (ORG - p0x) / dsx, ax1 = (1.0f - p0x) / dsx;
    const float ay0 = (ORG - p0y) / dsy, ay1 = (1.0f - p0y) / dsy;
    const float az0 = (ORG - p0z) / dsz, az1 = (1.0f - p0z) / dsz;
    const float amin = fmaxf(fmaxf(fminf(ax0, ax1), fminf(ay0, ay1)),
                             fmaxf(fminf(az0, az1), 0.0f));
    const float amax = fminf(fminf(fmaxf(ax0, ax1), fmaxf(ay0, ay1)),
                             fminf(fmaxf(az0, az1), 1.0f));

    float result = 0.0f;

    if (amax > amin) {
        const float rlen = sqrtf(dx * dx + dy * dy + dz * dz);

        // entry point in voxel units (use true d, as the reference does)
        const float tx = (p0x + amin * dx - ORG) * INV;
        const float ty = (p0y + amin * dy - ORG) * INV;
        const float tz = (p0z + amin * dz - ORG) * INV;

        const int sx = (dsx > 0.0f) ? 1 : -1;
        const int sy = (dsy > 0.0f) ? 1 : -1;
        const int sz = (dsz > 0.0f) ? 1 : -1;

        // next plane crossed per axis, in alpha units
        const float jx = (dsx > 0.0f) ? floorf(tx) + 1.0f : ceilf(tx) - 1.0f;
        const float jy = (dsy > 0.0f) ? floorf(ty) + 1.0f : ceilf(ty) - 1.0f;
        const float jz = (dsz > 0.0f) ? floorf(tz) + 1.0f : ceilf(tz) - 1.0f;

        float axN = fmaxf((ORG + jx * VOX - p0x) / dsx, amin);
        float ayN = fmaxf((ORG + jy * VOX - p0y) / dsy, amin);
        float azN = fmaxf((ORG + jz * VOX - p0z) / dsz, amin);
        const float dax = VOX / fabsf(dsx);
        const float day = VOX / fabsf(dsy);
        const float daz = VOX / fabsf(dsz);

        // initial voxel from midpoint of first segment (robust on boundaries,
        // matches the reference's midpoint indexing)
        const float aF = fminf(fminf(axN, ayN), fminf(azN, amax));
        const float am = 0.5f * (amin + aF);
        int ix = (int)floorf((p0x + am * dx - ORG) * INV);
        int iy = (int)floorf((p0y + am * dy - ORG) * INV);
        int iz = (int)floorf((p0z + am * dz - ORG) * INV);

        // flat voxel index maintained incrementally
        int flat = (ix * GRID_N + iy) * GRID_N + iz;
        const int fsx = sx * GRID_N * GRID_N;   // +/- 16384
        const int fsy = sy * GRID_N;            // +/- 128
        const int fsz = sz;                     // +/- 1

        // warm L0/L2 for the first gather (global_prefetch_b8)
        {
            int cx = ix < 0 ? 0 : (ix > 127 ? 127 : ix);
            int cy = iy < 0 ? 0 : (iy > 127 ? 127 : iy);
            int cz = iz < 0 ? 0 : (iz > 127 ? 127 : iz);
            __builtin_prefetch(image + ((cx * GRID_N + cy) * GRID_N + cz), 0, 3);
        }

        float alpha = amin;
        #pragma unroll 1
        for (int it = 0; it < 3 * GRID_N + 4; ++it) {
            const float aN  = fminf(fminf(axN, ayN), fminf(azN, amax));
            const float seg = aN - alpha;            // provably >= 0

            // single fused in-bounds test: all of ix,iy,iz in [0,128)
            const unsigned ubits = (unsigned)ix | (unsigned)iy | (unsigned)iz;
            const bool inb = (ubits < 128u);

            // branchless clamped gather + masked accumulate
            const int   flatc = inb ? flat : 0;
            const float w     = inb ? seg  : 0.0f;
            result = fmaf(w, image[flatc], result);

            if (aN >= amax) break;

            if (axN <= aN) { ix += sx; flat += fsx; axN += dax; }
            if (ayN <= aN) { iy += sy; flat += fsy; ayN += day; }
            if (azN <= aN) { iz += sz; flat += fsz; azN += daz; }
            alpha = aN;
        }
        result *= rlen;   // segments accumulated in alpha units
    }

    out[ray] = result;
}

extern "C" void kernel_launch(void* const* d_in, const int* in_sizes, int n_in,
                              void* d_out, int out_size, void* d_ws, size_t ws_size,
                              hipStream_t stream) {
    const float* image = (const float*)d_in[0];   // 128^3 float32
    const float* lors  = (const float*)d_in[1];   // [N,6] float32
    float* out = (float*)d_out;                   // [N] float32
    const int n = in_sizes[1] / 6;

    const int blocks = (n + THREADS - 1) / THREADS;
    siddon_fp_kernel<<<blocks, THREADS, 0, stream>>>(image, lors, out, n);
}